// LinearSpline_61057255080647
// MI455X (gfx1250) — compile-verified
//
#include <hip/hip_runtime.h>

// LinearSpline activation for MI455X (gfx1250), bandwidth-bound streaming kernel.
// x: [16, 64, 256, 256] f32; per-channel 129-knot B1-spline table.
// No matmul -> no WMMA. CDNA5 paths used: async global->LDS table staging,
// LDS gathers, non-temporal B128 streaming loads/stores, wave32 blocks.

#define NUM_ACT      64
#define TAB_SIZE     129
#define RANGE_F      4.0f
#define GRID_F       0.0625f      // 2*RANGE/(SIZE-1)
#define INV_GRID_F   16.0f
#define HALF_GRID_F  0.03125f
#define PLANE        65536        // 256*256 elements per (batch, channel)
#define CHUNK        16384        // elements per block
#define BLK_PER_PLANE (PLANE / CHUNK)   // 4
#define THREADS      256          // 8 wave32s

typedef float v4f __attribute__((ext_vector_type(4)));
typedef __attribute__((address_space(1))) int  gint_t;
typedef __attribute__((address_space(3))) int  lint_t;

__device__ __forceinline__ float spline_eval(float xv, float s, float inv_s,
                                             const float* __restrict__ tab) {
    float xg   = xv * s;
    float xc   = fminf(fmaxf(xg, -RANGE_F), RANGE_F - GRID_F);
    float fl   = floorf(xc * INV_GRID_F);             // bin index in [-64, 63]
    float frac = fmaf(xg, INV_GRID_F, -fl);           // xg/GRID - floored
    int   k    = (int)fl + (TAB_SIZE / 2);            // [0, 127]
    float c0   = tab[k];                              // ds_load
    float c1   = tab[k + 1];                          // ds_load
    return (fmaf(frac, c1 - c0, c0) - HALF_GRID_F) * inv_s;
}

__global__ __launch_bounds__(THREADS) void
linear_spline_kernel(const float* __restrict__ x,
                     const float* __restrict__ coeffs,    // [NUM_ACT * TAB_SIZE]
                     const float* __restrict__ scaling,   // [NUM_ACT]
                     float* __restrict__ out) {
    __shared__ float tab[TAB_SIZE + 3];   // pad; 129 coefficients of this channel

    const int bid = blockIdx.x;
    const int tid = threadIdx.x;
    const int c   = (bid / BLK_PER_PLANE) & (NUM_ACT - 1);   // channel, block-uniform

    const float* __restrict__ ctab = coeffs + c * TAB_SIZE;

    // ---- Stage the 129-entry channel table into LDS (gfx1250 async-copy path) ----
    if (tid < TAB_SIZE) {
#if defined(__gfx1250__) && __has_builtin(__builtin_amdgcn_global_load_async_to_lds_b32)
        __builtin_amdgcn_global_load_async_to_lds_b32(
            (gint_t*)(ctab + tid),
            (lint_t*)(tab + tid),
            /*offset=*/0, /*cpol=*/0);
#else
        tab[tid] = ctab[tid];
#endif
    }
#if defined(__gfx1250__) && __has_builtin(__builtin_amdgcn_global_load_async_to_lds_b32)
#if __has_builtin(__builtin_amdgcn_s_wait_asynccnt)
    __builtin_amdgcn_s_wait_asynccnt(0);
#else
    asm volatile("s_wait_asynccnt 0" ::: "memory");
#endif
#endif
    __syncthreads();

    const float s     = scaling[c];      // block-uniform -> scalar load
    const float inv_s = 1.0f / s;

    const size_t base = (size_t)bid * CHUNK;
    const v4f* __restrict__ xin = (const v4f*)(x + base);
    v4f* __restrict__       pot = (v4f*)(out + base);

    // CHUNK/4 = 4096 float4 per block; 256 threads -> 16 iterations
    constexpr int ITERS = CHUNK / (THREADS * 4);
#pragma unroll 4
    for (int it = 0; it < ITERS; ++it) {
        const int i = it * THREADS + tid;
        v4f v = __builtin_nontemporal_load(xin + i);   // streaming B128 load (TH=NT)
        v4f r;
        r.x = spline_eval(v.x, s, inv_s, tab);
        r.y = spline_eval(v.y, s, inv_s, tab);
        r.z = spline_eval(v.z, s, inv_s, tab);
        r.w = spline_eval(v.w, s, inv_s, tab);
        __builtin_nontemporal_store(r, pot + i);       // streaming B128 store (TH=NT)
    }
}

extern "C" void kernel_launch(void* const* d_in, const int* in_sizes, int n_in,
                              void* d_out, int out_size, void* d_ws, size_t ws_size,
                              hipStream_t stream) {
    const float* x       = (const float*)d_in[0];   // [16*64*256*256] f32
    const float* coeffs  = (const float*)d_in[1];   // [64*129] f32
    const float* scaling = (const float*)d_in[2];   // [64] f32
    // d_in[3] (zero_knot_indexes) is analytically c*129 + 64; recomputed in-kernel.
    (void)in_sizes; (void)n_in; (void)d_ws; (void)ws_size;

    const int total  = out_size;            // 67,108,864
    const int nblock = total / CHUNK;       // 4096
    linear_spline_kernel<<<nblock, THREADS, 0, stream>>>(x, coeffs, scaling,
                                                         (float*)d_out);
}